// GCNLayer_44770739094190
// MI455X (gfx1250) — compile-verified
//
#include <hip/hip_runtime.h>

// ---------------------------------------------------------------------------
// SpMM (GCN aggregate) for MI455X / gfx1250, wave32.
//
// Roofline: 205 MFLOP vs ~70 MB unique traffic (x table is L2-resident in the
// 192MB L2) -> memory bound. Strategy:
//   * one wave per chunk of 16 sorted-by-row edges
//   * gather 16 neighbor rows of x (coalesced 64B half-wave segments, L2 hits)
//   * densify the segment reduction as S(16x16, one-hot*val) x B(16x64)
//     with V_WMMA_F32_16X16X4_F32 (f32 end-to-end, matches reference dtype)
//   * only ~nrows (~2 avg, rows sorted) rows/chunk leave via f32 atomics
//
// v2: all cross-lane shuffles hoisted out of conditionals (branch-free
//     v_cndmask instead of exec-masked ds_bpermute blocks), (rloc|col) packed
//     into one shuffle, wave-uniform early-out on the writeback loop.
// ---------------------------------------------------------------------------

typedef __attribute__((ext_vector_type(2))) float v2f;
typedef __attribute__((ext_vector_type(8))) float v8f;

#define D_FEAT 64

__global__ void __launch_bounds__(256) zero_kernel(float* __restrict__ out, int n) {
    int i = blockIdx.x * blockDim.x + threadIdx.x;
    if (i < n) out[i] = 0.0f;
}

__global__ void __launch_bounds__(256) spmm_wmma_kernel(
        const int*   __restrict__ rows,
        const int*   __restrict__ cols,
        const float* __restrict__ vals,
        const float* __restrict__ x,
        float*       __restrict__ out,
        int nEdges, int nChunks) {

    const int lane  = threadIdx.x & 31;
    const int hi    = lane >> 4;     // 0: lanes 0-15, 1: lanes 16-31
    const int nidx  = lane & 15;     // M index for A / N index for B and C/D
    const int wavesPerBlock = blockDim.x >> 5;
    const int wave  = blockIdx.x * wavesPerBlock + (threadIdx.x >> 5);
    const int nWaves = gridDim.x * wavesPerBlock;

    for (int chunk = wave; chunk < nChunks; chunk += nWaves) {
        const int ebase  = chunk * 16;
        const int nvalid = min(16, nEdges - ebase);

        // --- lanes 0..15 each own one edge of the chunk -------------------
        int   myRow = 0x7fffffff;
        int   myCol = 0;
        float myVal = 0.0f;
        if (lane < nvalid) {
            myRow = rows[ebase + lane];
            myCol = cols[ebase + lane];
            myVal = vals[ebase + lane];
        }

        // --- segment boundaries -> compacted local row index (rank) ------
        const int prevRow = __shfl(myRow, (lane == 0) ? 0 : (lane - 1));
        const bool bnd = (lane > 0) && (lane < nvalid) && (myRow != prevRow);
        const unsigned mask = (unsigned)__ballot(bnd);           // bits 1..15
        const int rloc  = __popc(mask & (unsigned)((2ull << lane) - 1ull));
        const int nrows = __shfl(rloc, nvalid - 1) + 1;          // distinct rows

        // --- local row -> global row map: lane m holds grow[m] ------------
        // (colliding writers carry identical row ids; unwritten lanes are
        //  m >= nrows and never read)
        const int dst = (lane < nvalid) ? rloc : lane;
        const int growReg = __builtin_amdgcn_ds_permute(dst * 4, myRow);

        // one shuffle carries both the local-row rank and the gather column
        const int packed = (rloc << 24) | myCol;                 // col < 2^24

        // --- accumulate S(16x16) x B(16x64) with WMMA f32 16x16x4 ---------
        v8f acc0 = {}, acc1 = {}, acc2 = {}, acc3 = {};

        #pragma unroll
        for (int t = 0; t < 4; ++t) {
            // this lane supplies K = k0, k1 of the current 4-wide K-step
            const int k0 = 4 * t + (hi ? 2 : 0);
            const int k1 = k0 + 1;

            // unconditional cross-lane reads -> branch-free selects
            const int   p0  = __shfl(packed, k0);
            const int   p1  = __shfl(packed, k1);
            const float sv0 = __shfl(myVal, k0);
            const float sv1 = __shfl(myVal, k1);

            // A fragment: A[m][k] = (rloc[k] == m) ? vals[k] : 0
            // (invalid edges carry myVal == 0 -> contribute nothing)
            v2f a;
            a.x = ((p0 >> 24) == nidx) ? sv0 : 0.0f;
            a.y = ((p1 >> 24) == nidx) ? sv1 : 0.0f;

            // gathered neighbor rows for this K-step (L2-resident)
            const int c0 = p0 & 0x00FFFFFF;
            const int c1 = p1 & 0x00FFFFFF;
            const float* xr0 = x + (size_t)c0 * D_FEAT + nidx;
            const float* xr1 = x + (size_t)c1 * D_FEAT + nidx;

            // 4 feature tiles of 16 columns each
            {
                v2f b; b.x = xr0[0];  b.y = xr1[0];
                acc0 = __builtin_amdgcn_wmma_f32_16x16x4_f32(
                        false, a, false, b, (short)0, acc0, false, false);
            }
            {
                v2f b; b.x = xr0[16]; b.y = xr1[16];
                acc1 = __builtin_amdgcn_wmma_f32_16x16x4_f32(
                        false, a, false, b, (short)0, acc1, false, false);
            }
            {
                v2f b; b.x = xr0[32]; b.y = xr1[32];
                acc2 = __builtin_amdgcn_wmma_f32_16x16x4_f32(
                        false, a, false, b, (short)0, acc2, false, false);
            }
            {
                v2f b; b.x = xr0[48]; b.y = xr1[48];
                acc3 = __builtin_amdgcn_wmma_f32_16x16x4_f32(
                        false, a, false, b, (short)0, acc3, false, false);
            }
        }

        // --- writeback: C/D layout VGPR v -> M = v + 8*hi, N = nidx -------
        // nrows is wave-uniform: if v >= nrows then both halves (m = v and
        // m = v+8) are past the last live row -> uniform early-out.
        #pragma unroll
        for (int v = 0; v < 8; ++v) {
            if (v >= nrows) break;                       // uniform s_cbranch
            const int m = v + (hi ? 8 : 0);
            const int grow = __shfl(growReg, m);         // unconditional
            if (m < nrows) {                             // per-lane predicate
                float* o = out + (size_t)grow * D_FEAT + nidx;
                atomicAdd(o + 0,  acc0[v]);
                atomicAdd(o + 16, acc1[v]);
                atomicAdd(o + 32, acc2[v]);
                atomicAdd(o + 48, acc3[v]);
            }
        }
    }
}

extern "C" void kernel_launch(void* const* d_in, const int* in_sizes, int n_in,
                              void* d_out, int out_size, void* d_ws, size_t ws_size,
                              hipStream_t stream) {
    const int*   adj_rows = (const int*)  d_in[0];
    const int*   adj_cols = (const int*)  d_in[1];
    const float* adj_vals = (const float*)d_in[2];
    const float* x        = (const float*)d_in[3];
    float*       out      = (float*)      d_out;

    const int E = in_sizes[0];

    // Atomics accumulate across graph replays: zero d_out every launch.
    zero_kernel<<<(out_size + 255) / 256, 256, 0, stream>>>(out, out_size);

    const int nChunks = (E + 15) / 16;               // 100000 for reference sizes
    const int threads = 256;                          // 8 waves / block
    int blocks = 2048;                                // grid-stride over chunks
    int maxBlocks = (nChunks + 7) / 8;
    if (blocks > maxBlocks) blocks = maxBlocks;
    if (blocks < 1) blocks = 1;

    spmm_wmma_kernel<<<blocks, threads, 0, stream>>>(
        adj_rows, adj_cols, adj_vals, x, out, E, nChunks);
}